// SlotConditionedReadModel_3204045603514
// MI455X (gfx1250) — compile-verified
//
#include <hip/hip_runtime.h>
#include <math.h>

// ---------------------------------------------------------------------------
// SlotConditionedReadModel, MI455X (gfx1250) implementation.
//
// Key observation: the encoder (embed -> FF -> residual -> LayerNorm) is
// pointwise in the *token id* (VOCAB=64), so h[b,l,:] takes only 64 distinct
// values. We precompute per-token-id tables once with fp32 WMMA
// (v_wmma_f32_16x16x4_f32), reduce top_k to a per-row histogram + global
// score ranking, and the per-batch attention to table lookups.
// ---------------------------------------------------------------------------

#define Hdim   64
#define H2dim  128
#define NSLOT  8
#define EPSV   1e-5f

typedef __attribute__((ext_vector_type(2))) float v2f;
typedef __attribute__((ext_vector_type(8))) float v8f;

// Compute one 16x16 fp32 tile of  A[64xK] @ B  where B[k][n] = W[n*ldw + k]
// (i.e. the "x @ W^T" pattern). A is row-major in LDS, W row-major in global.
// WMMA f32 16x16x4 operand layouts (ISA 7.12.2):
//   A 16x4 : lanes 0-15 -> K=0,1 in v[0],v[1]; lanes 16-31 -> K=2,3
//   B 4x16 : lanes 0-15 -> K=0,1 rows (N=lane); lanes 16-31 -> K=2,3
//   C 16x16: VGPR r -> M=r (lanes 0-15), M=r+8 (lanes 16-31), N = lane&15
__device__ __forceinline__ v8f wmma_tile_f32(const float* __restrict__ A, int lda,
                                             const float* __restrict__ W, int ldw,
                                             int m0, int n0, int K, int lane)
{
    v8f c = {0.f, 0.f, 0.f, 0.f, 0.f, 0.f, 0.f, 0.f};
    const int l15 = lane & 15;
    const int kh  = (lane >> 4) << 1;          // 0 for lanes 0-15, 2 for 16-31
    const float* ar = A + (m0 + l15) * lda;
    const float* wr = W + (n0 + l15) * ldw;
    for (int k0 = 0; k0 < K; k0 += 4) {
        v2f a, b;
        a[0] = ar[k0 + kh];
        a[1] = ar[k0 + kh + 1];
        b[0] = wr[k0 + kh];
        b[1] = wr[k0 + kh + 1];
        c = __builtin_amdgcn_wmma_f32_16x16x4_f32(false, a, false, b,
                                                  (short)0, c, false, false);
    }
    return c;
}

// ---------------------------------------------------------------------------
// Kernel 1: one workgroup (256 threads = 8 wave32). Builds all tables.
// ---------------------------------------------------------------------------
__global__ __launch_bounds__(256) void build_tables_kernel(
    const float* __restrict__ embed,
    const float* __restrict__ w1, const float* __restrict__ b1,
    const float* __restrict__ w2, const float* __restrict__ b2,
    const float* __restrict__ gamma, const float* __restrict__ beta,
    const float* __restrict__ gate_w, const float* __restrict__ gate_b,
    const float* __restrict__ qi_w, const float* __restrict__ qi_b,
    const float* __restrict__ sk_w, const float* __restrict__ sk_b,
    const float* __restrict__ qr_w, const float* __restrict__ qr_b,
    const float* __restrict__ out_w, const float* __restrict__ out_b,
    float* __restrict__ h_table, float* __restrict__ sk_table,
    float* __restrict__ qi_table, float* __restrict__ qr_table,
    float* __restrict__ out_table, float* __restrict__ score_table,
    int* __restrict__ rank_order)
{
    __shared__ float sA[Hdim * Hdim];    // h0, later reused as h_table
    __shared__ float sT[Hdim * H2dim];   // relu intermediate
    __shared__ float sScore[Hdim];

    const int tid  = threadIdx.x;
    const int wave = tid >> 5;
    const int lane = tid & 31;

    // Stage 0: load embed (the 64 possible h0 rows) into LDS.
    for (int i = tid; i < Hdim * Hdim; i += 256) sA[i] = embed[i];
    __syncthreads();

    // Stage 1: sT[64][128] = relu(sA @ w1^T + b1).  32 tiles, 4 per wave.
    for (int i = 0; i < 4; ++i) {
        const int t  = wave + i * 8;          // 0..31
        const int m0 = (t >> 3) << 4;
        const int n0 = (t & 7) << 4;
        v8f c = wmma_tile_f32(sA, Hdim, w1, Hdim, m0, n0, Hdim, lane);
        const int n  = n0 + (lane & 15);
        const int mb = m0 + ((lane >> 4) << 3);
        const float bn = b1[n];
#pragma unroll
        for (int r = 0; r < 8; ++r) {
            float v = c[r] + bn;
            sT[(mb + r) * H2dim + n] = v > 0.f ? v : 0.f;
        }
    }
    __syncthreads();

    // Stage 2: sA[64][64] = sT @ w2^T + b2 + h0 (residual re-read from global).
    for (int i = 0; i < 2; ++i) {
        const int t  = wave + i * 8;          // 0..15
        const int m0 = (t >> 2) << 4;
        const int n0 = (t & 3) << 4;
        v8f c = wmma_tile_f32(sT, H2dim, w2, H2dim, m0, n0, H2dim, lane);
        const int n  = n0 + (lane & 15);
        const int mb = m0 + ((lane >> 4) << 3);
        const float bn = b2[n];
#pragma unroll
        for (int r = 0; r < 8; ++r)
            sA[(mb + r) * Hdim + n] = c[r] + bn + embed[(mb + r) * Hdim + n];
    }
    __syncthreads();

    // Stage 3: LayerNorm in place (one thread per token id), gate score.
    if (tid < Hdim) {
        float* row = sA + tid * Hdim;
        float mu = 0.f;
        for (int j = 0; j < Hdim; ++j) mu += row[j];
        mu *= (1.f / 64.f);
        float var = 0.f;
        for (int j = 0; j < Hdim; ++j) { float d = row[j] - mu; var += d * d; }
        var *= (1.f / 64.f);
        const float rs = rsqrtf(var + EPSV);
        float sc = 0.f;
        for (int j = 0; j < Hdim; ++j) {
            float hn = (row[j] - mu) * rs * gamma[j] + beta[j];
            row[j] = hn;
            h_table[tid * Hdim + j] = hn;
            sc += hn * gate_w[j];
        }
        sc += gate_b[0];
        sScore[tid] = sc;
        score_table[tid] = sc;
    }
    __syncthreads();

    // Stage 4: global descending ranking of the 64 scores (batch independent).
    if (tid < Hdim) {
        const float my = sScore[tid];
        int rank = 0;
        for (int u = 0; u < Hdim; ++u) {
            const float su = sScore[u];
            if (su > my || (su == my && u < tid)) ++rank;
        }
        rank_order[rank] = tid;
    }
    __syncthreads();

    // Stage 5: four projection tables, biases baked in
    // (qr_b / out_b are legal to bake since the attention weights sum to 1).
    const float* Wm[4] = { sk_w, qi_w, qr_w, out_w };
    const float* Bv[4] = { sk_b, qi_b, qr_b, out_b };
    float*       Om[4] = { sk_table, qi_table, qr_table, out_table };
    for (int i = 0; i < 8; ++i) {
        const int tt  = wave + i * 8;          // 0..63 tiles (4 mats x 16)
        const int mat = tt >> 4;
        const int t   = tt & 15;
        const int m0  = (t >> 2) << 4;
        const int n0  = (t & 3) << 4;
        v8f c = wmma_tile_f32(sA, Hdim, Wm[mat], Hdim, m0, n0, Hdim, lane);
        const int n  = n0 + (lane & 15);
        const int mb = m0 + ((lane >> 4) << 3);
        const float bn = Bv[mat][n];
        float* O = Om[mat];
#pragma unroll
        for (int r = 0; r < 8; ++r)
            O[(mb + r) * Hdim + n] = c[r] + bn;
    }
}

// ---------------------------------------------------------------------------
// Kernel 2: one block per batch row. Histogram -> slot fill -> attention.
// ---------------------------------------------------------------------------
__device__ __forceinline__ void softmax8_inplace(volatile float* lg, float* att)
{
    float mx = lg[0];
    for (int s = 1; s < NSLOT; ++s) mx = fmaxf(mx, lg[s]);
    float e[NSLOT];
    float sum = 0.f;
    for (int s = 0; s < NSLOT; ++s) { e[s] = expf(lg[s] - mx); sum += e[s]; }
    const float inv = 1.f / sum;
    for (int s = 0; s < NSLOT; ++s) att[s] = e[s] * inv;
}

__global__ __launch_bounds__(256) void select_attend_kernel(
    const int* __restrict__ seq, int L,
    const float* __restrict__ h_table,  const float* __restrict__ sk_table,
    const float* __restrict__ qi_table, const float* __restrict__ qr_table,
    const float* __restrict__ out_table,
    const int* __restrict__ rank_order,
    float* __restrict__ out)
{
    __shared__ unsigned cnt[Hdim];
    __shared__ int   slots[NSLOT];
    __shared__ int   last_v;
    __shared__ float qinit[Hdim];
    __shared__ float qref[Hdim];
    __shared__ float lg[NSLOT];
    __shared__ float att[NSLOT];

    const int b   = blockIdx.x;
    const int tid = threadIdx.x;
    const int* row = seq + (size_t)b * L;

    if (tid < Hdim) cnt[tid] = 0u;
    if (tid == 0)   last_v = row[L - 1] & (Hdim - 1);
    __syncthreads();

    // Histogram of token ids (LDS ds_add atomics; integer -> deterministic).
    for (int t = tid; t < L; t += 256)
        atomicAdd(&cnt[row[t] & (Hdim - 1)], 1u);
    __syncthreads();

    // top_k(scores, 8): walk ids in descending-score order, emit each id
    // count times (ties within an id map to identical h_table rows, matching
    // jax.lax.top_k's earliest-index tie-break semantics).
    if (tid == 0) {
        int s = 0;
        for (int r = 0; r < Hdim && s < NSLOT; ++r) {
            const int v = rank_order[r];
            unsigned c = cnt[v];
            while (c-- && s < NSLOT) slots[s++] = v;
        }
        while (s < NSLOT) slots[s++] = 0;   // unreachable for L >= 8
    }
    __syncthreads();

    // q_init = qi_table[last token id]   (qi_b baked in)
    if (tid < Hdim) qinit[tid] = qi_table[last_v * Hdim + tid];
    __syncthreads();

    // Stage-1 logits: dot(sk_table[v_s], q_init) / sqrt(H)
    if (tid < NSLOT) {
        const float* k = sk_table + slots[tid] * Hdim;
        float d = 0.f;
        for (int j = 0; j < Hdim; ++j) d += k[j] * qinit[j];
        lg[tid] = d * 0.125f;
    }
    __syncthreads();
    if (tid == 0) softmax8_inplace(lg, att);
    __syncthreads();

    // q_refined = q_init + sum_s r_attn[s] * qr_table[v_s]  (qr_b baked)
    if (tid < Hdim) {
        float q = qinit[tid];
        for (int s = 0; s < NSLOT; ++s)
            q += att[s] * qr_table[slots[s] * Hdim + tid];
        qref[tid] = q;
    }
    __syncthreads();

    // Stage-2 logits: dot(h_table[v_s], q_refined) / sqrt(H)
    if (tid < NSLOT) {
        const float* m = h_table + slots[tid] * Hdim;
        float d = 0.f;
        for (int j = 0; j < Hdim; ++j) d += m[j] * qref[j];
        lg[tid] = d * 0.125f;
    }
    __syncthreads();
    if (tid == 0) softmax8_inplace(lg, att);
    __syncthreads();

    // out = sum_s attn[s] * out_table[v_s]   (out_b baked)
    if (tid < Hdim) {
        float o = 0.f;
        for (int s = 0; s < NSLOT; ++s)
            o += att[s] * out_table[slots[s] * Hdim + tid];
        out[(size_t)b * Hdim + tid] = o;
    }
}

// ---------------------------------------------------------------------------
extern "C" void kernel_launch(void* const* d_in, const int* in_sizes, int n_in,
                              void* d_out, int out_size, void* d_ws, size_t ws_size,
                              hipStream_t stream)
{
    const int*   seq    = (const int*)d_in[0];
    const float* embed  = (const float*)d_in[1];
    const float* w1     = (const float*)d_in[2];
    const float* b1     = (const float*)d_in[3];
    const float* w2     = (const float*)d_in[4];
    const float* b2     = (const float*)d_in[5];
    const float* gamma  = (const float*)d_in[6];
    const float* beta   = (const float*)d_in[7];
    const float* gate_w = (const float*)d_in[8];
    const float* gate_b = (const float*)d_in[9];
    const float* qi_w   = (const float*)d_in[10];
    const float* qi_b   = (const float*)d_in[11];
    const float* sk_w   = (const float*)d_in[12];
    const float* sk_b   = (const float*)d_in[13];
    const float* qr_w   = (const float*)d_in[14];
    const float* qr_b   = (const float*)d_in[15];
    const float* out_w  = (const float*)d_in[16];
    const float* out_b  = (const float*)d_in[17];
    float* out = (float*)d_out;

    const int B = out_size / Hdim;          // 1024
    const int L = in_sizes[0] / B;          // 2048

    // Workspace layout (floats): 5 tables of 64x64, 64 scores, 64 ranks.
    float* ws          = (float*)d_ws;
    float* h_table     = ws + 0 * 4096;
    float* sk_table    = ws + 1 * 4096;
    float* qi_table    = ws + 2 * 4096;
    float* qr_table    = ws + 3 * 4096;
    float* out_table   = ws + 4 * 4096;
    float* score_table = ws + 5 * 4096;
    int*   rank_order  = (int*)(ws + 5 * 4096 + 64);

    build_tables_kernel<<<1, 256, 0, stream>>>(
        embed, w1, b1, w2, b2, gamma, beta, gate_w, gate_b,
        qi_w, qi_b, sk_w, sk_b, qr_w, qr_b, out_w, out_b,
        h_table, sk_table, qi_table, qr_table, out_table,
        score_table, rank_order);

    select_attend_kernel<<<B, 256, 0, stream>>>(
        seq, L, h_table, sk_table, qi_table, qr_table, out_table,
        rank_order, out);
}